// STCA_TCA_Loss_80504866996734
// MI455X (gfx1250) — compile-verified
//
#include <hip/hip_runtime.h>
#include <stdint.h>

// STCA/TCA loss on MI455X (gfx1250): streaming, bandwidth-bound.
// Traffic floor: 128 MB of vmem @ 23.3 TB/s ~= 5.5 us (vlastmem is dead -> never read).
// One thread per row; async global->LDS double-buffered staging (ASYNCcnt path).

#define CGAP 10
#define NEGF (-1.0e30f)
#define ROWS_PER_BLOCK 128
#define T_CHUNK 32
#define ROW_STRIDE (T_CHUNK + 4)   // floats: 36 -> 144B row stride, 16B aligned, bank-friendly

template <int K>
__device__ __forceinline__ void wait_asynccnt() {
  asm volatile("s_wait_asynccnt %0" :: "n"(K) : "memory");
}

// GVS mode: mem_addr = SGPR64 base + VGPR32 byte offset; LDS dest byte addr in VGPR.
__device__ __forceinline__ void async_load_b128_to_lds(uint32_t lds_byte,
                                                       const float* base,
                                                       uint32_t byte_off) {
  asm volatile("global_load_async_to_lds_b128 %0, %1, %2"
               :: "v"(lds_byte), "v"(byte_off), "s"(base)
               : "memory");
}

__device__ __forceinline__ void stage_chunk(const float* __restrict__ vmem,
                                            int row0, int tBase, int T,
                                            uint32_t ldsBase, int tid) {
  const int q    = tid & 7;   // 16B quad within the 32-timestep chunk
  const int rsub = tid >> 3;  // 0..15
#pragma unroll
  for (int i = 0; i < 8; ++i) {       // 8 async-load instructions per wave per chunk
    const int r = rsub + i * 16;      // local row 0..127
    const long long elem = (long long)(row0 + r) * T + tBase + q * 4;
    const uint32_t gofs = (uint32_t)(elem * 4);                        // < 2^31 for this problem
    const uint32_t lofs = ldsBase + (uint32_t)(r * (ROW_STRIDE * 4) + q * 16);
    async_load_b128_to_lds(lofs, vmem, gofs);
  }
}

extern "C" __global__ void __launch_bounds__(ROWS_PER_BLOCK)
stca_main(const float* __restrict__ vmem, const int* __restrict__ labels,
          float* __restrict__ spike_out, float* __restrict__ block_partials,
          int N, int T) {
  __shared__ float buf[2][ROWS_PER_BLOCK * ROW_STRIDE];
  __shared__ float wsum[ROWS_PER_BLOCK / 32];

  const int tid  = threadIdx.x;
  const int row0 = blockIdx.x * ROWS_PER_BLOCK;
  const int row  = row0 + tid;

  const uint32_t ldsBase0 = (uint32_t)(uintptr_t)&buf[0][0];
  const uint32_t ldsBase1 = (uint32_t)(uintptr_t)&buf[1][0];

  // O(1) sequential-scan state for this row's cluster decomposition
  float max_all = NEGF;           // max v over whole row
  float gap_max = NEGF;           // max v since last spike (exclusive) up to current t
  float cur_max = NEGF;           // max v over current cluster span [first..last spike]
  float best_max = 0.0f;          // max v over smallest-so-far cluster span
  int last_spike = -(CGAP + 2);
  int cur_cnt = 0;
  int best_cnt = 0x7fffffff;      // strict '<' on close keeps earliest cluster on ties
  int num_cl = 0;

  const int nchunk = T / T_CHUNK;

  stage_chunk(vmem, row0, 0, T, ldsBase0, tid);

  for (int c = 0; c < nchunk; ++c) {
    if (c + 1 < nchunk) {
      stage_chunk(vmem, row0, (c + 1) * T_CHUNK, T,
                  ((c + 1) & 1) ? ldsBase1 : ldsBase0, tid);
      wait_asynccnt<8>();   // in-order: chunk c's 8 loads have completed
    } else {
      wait_asynccnt<0>();
    }
    __syncthreads();

    const float* rowbuf = &buf[c & 1][tid * ROW_STRIDE];
#pragma unroll
    for (int q = 0; q < T_CHUNK / 4; ++q) {
      const float4 w = *(const float4*)(rowbuf + 4 * q);
      const int tb = c * T_CHUNK + 4 * q;
      const float vv[4] = {w.x, w.y, w.z, w.w};
#pragma unroll
      for (int k = 0; k < 4; ++k) {
        const float v = vv[k];
        const int t = tb + k;
        const float g = fmaxf(gap_max, v);          // max over (last_spike, t]
        max_all = fmaxf(max_all, v);
        const bool sp   = (v >= 0.0f);
        const bool newc = sp && ((t - last_spike) > CGAP);
        const bool cont = sp && !newc;
        const bool close = newc && (cur_cnt > 0) && (cur_cnt < best_cnt);
        best_cnt = close ? cur_cnt : best_cnt;
        best_max = close ? cur_max : best_max;
        num_cl  += newc ? 1 : 0;
        cur_max  = newc ? v : (cont ? fmaxf(cur_max, g) : cur_max);
        cur_cnt  = newc ? 1 : (cur_cnt + (cont ? 1 : 0));
        last_spike = sp ? t : last_spike;
        gap_max    = sp ? NEGF : g;
      }
    }
    __syncthreads();
  }

  // close final cluster
  if (cur_cnt > 0 && cur_cnt < best_cnt) best_max = cur_max;

  const int b = row / N;
  const int n = row - b * N;
  const float target = (labels[b] == n) ? 1.0f : 0.0f;
  const float nc = (float)num_cl;
  spike_out[row] = nc;

  float contrib = 0.0f;
  if (target > nc)      contrib = -max_all;   // target==1, no spikes
  else if (target < nc) contrib = best_max;   // max v over smallest cluster's span

  // deterministic block reduction: fixed wave32 shuffle tree + fixed-order sum
#pragma unroll
  for (int off = 16; off > 0; off >>= 1)
    contrib += __shfl_down(contrib, off, 32);
  if ((tid & 31) == 0) wsum[tid >> 5] = contrib;
  __syncthreads();
  if (tid == 0) {
    float s = 0.0f;
#pragma unroll
    for (int w = 0; w < ROWS_PER_BLOCK / 32; ++w) s += wsum[w];
    block_partials[blockIdx.x] = s;
  }
}

extern "C" __global__ void stca_reduce(const float* __restrict__ partials, int n,
                                       float* __restrict__ out) {
  if (threadIdx.x == 0 && blockIdx.x == 0) {
    float s = 0.0f;
    for (int i = 0; i < n; ++i) s += partials[i];  // fixed order -> deterministic
    out[0] = s;
  }
}

extern "C" void kernel_launch(void* const* d_in, const int* in_sizes, int n_in,
                              void* d_out, int out_size, void* d_ws, size_t ws_size,
                              hipStream_t stream) {
  (void)n_in; (void)ws_size;
  const float* vmem   = (const float*)d_in[0];   // [B,N,T] f32
  // d_in[1] (vlastmem) is unused by the reference forward -> never read (halves traffic)
  const int*   labels = (const int*)d_in[2];     // [B] i32
  float* out = (float*)d_out;                    // [0]=loss, [1..BN]=spike_output

  const int B  = in_sizes[2];
  const int BN = out_size - 1;
  const int N  = BN / B;
  const int T  = in_sizes[0] / BN;               // 1024; multiple of T_CHUNK

  float* partials = (float*)d_ws;
  const int blocks = BN / ROWS_PER_BLOCK;

  stca_main<<<blocks, ROWS_PER_BLOCK, 0, stream>>>(vmem, labels, out + 1, partials, N, T);
  stca_reduce<<<1, 32, 0, stream>>>(partials, blocks, out);
}